// STMBuffer_87935160418561
// MI455X (gfx1250) — compile-verified
//
#include <hip/hip_runtime.h>
#include <stdint.h>

// ---------------------------------------------------------------------------
// STM buffer update, fused single pass. Memory-bound: ~768 MB of traffic
// (read 256 MB buffer, write 256 MB decayed buf, write 256 MB ltm_ready)
// -> ~33 us at 23.3 TB/s. One wave32 per row; rows are double-buffered
// through LDS with gfx1250 GLOBAL_LOAD_ASYNC_TO_LDS_B128 (ASYNCcnt-tracked),
// consumed with ds_load_b128, written back with non-temporal stores.
// ---------------------------------------------------------------------------

#define B_ROWS 65536
#define COLS   1024
#define WAVES_PER_BLOCK 4          // 128 threads; 32 KB LDS/block (2 stages)

// clang native vector type so __builtin_nontemporal_* accepts it (b128 ops)
typedef float v4f __attribute__((ext_vector_type(4)));

// flat output offsets (float elements), reference tuple order:
// buf, occupancy, average_age, candidates, pressure, ltm_ready
static constexpr long long OUT_OCC  = 67108864LL;           // 65536*1024
static constexpr long long OUT_AGE  = 67108865LL;
static constexpr long long OUT_CAND = 67108866LL;
static constexpr long long OUT_PRES = 67174402LL;           // + 65536
static constexpr long long OUT_LTM  = 67174403LL;

// ---- init: input_strength * emotional_boost -> ws[0]; zero scalar accums ---
__global__ __launch_bounds__(32)
void stm_init_kernel(const float* __restrict__ new_input,
                     const float* __restrict__ valence_p,
                     float* __restrict__ ws,
                     float* __restrict__ out) {
  const int lane = threadIdx.x;            // single wave32
  float s = 0.0f;
  #pragma unroll 8
  for (int i = 0; i < 32; ++i)             // coalesced: i*32 + lane
    s += fabsf(new_input[i * 32 + lane]);
  #pragma unroll
  for (int o = 16; o; o >>= 1) s += __shfl_xor(s, o, 32);
  if (lane == 0) {
    const float input_strength = s * (1.0f / 1024.0f);
    const float boost = 1.0f + fabsf(valence_p[0]) * 0.5f;
    ws[0] = input_strength * boost;        // post-scatter strength at pos
    out[OUT_OCC]  = 0.0f;                  // zero atomic accumulators
    out[OUT_AGE]  = 0.0f;
    out[OUT_PRES] = 0.0f;
  }
}

// ---- main fused pass: one wave per row, async double-buffer through LDS ---
__global__ __launch_bounds__(32 * WAVES_PER_BLOCK)
void stm_main_kernel(const float* __restrict__ new_input,
                     const float* __restrict__ membuf,
                     const float* __restrict__ ages_in,
                     const float* __restrict__ tags_in,
                     const float* __restrict__ strengths_in,
                     const int*   __restrict__ wpos_p,
                     const float* __restrict__ valence_p,
                     const float* __restrict__ dt_p,
                     const float* __restrict__ ws,
                     float* __restrict__ out) {
  // [wave][stage][row]: 4 * 2 * 4KB = 32 KB static LDS
  __shared__ float lds_rows[WAVES_PER_BLOCK][2][COLS];

  const int lane        = threadIdx.x & 31;
  const int waveInBlock = threadIdx.x >> 5;
  const int totalWaves  = gridDim.x * WAVES_PER_BLOCK;
  const int w           = blockIdx.x * WAVES_PER_BLOCK + waveInBlock;

  const int   pos      = ((wpos_p[0] % B_ROWS) + B_ROWS) % B_ROWS;
  const float dt       = dt_p[0];
  const float valence  = valence_p[0];
  const float posStr   = ws[0];

  float ageSum = 0.0f, occCnt = 0.0f, candCnt = 0.0f;

  float* __restrict__ bufOut  = out;
  float* __restrict__ candOut = out + OUT_CAND;
  float* __restrict__ ltmOut  = out + OUT_LTM;

  // generic->LDS pointer: low 32 bits are the LDS byte address (ISA 10.2)
  const uint32_t ldsBase0 = (uint32_t)(uintptr_t)&lds_rows[waveInBlock][0][0];
  const uint32_t ldsBase1 = (uint32_t)(uintptr_t)&lds_rows[waveInBlock][1][0];

  // issue 8x global_load_async_to_lds_b128: 32 lanes x 16B x 8 = one 4KB row
  auto issue_row = [&](int r, uint32_t ldsBase) {
    const float* srcRow = (r == pos) ? new_input
                                     : membuf + (size_t)r * COLS;
    #pragma unroll
    for (int c = 0; c < 8; ++c) {
      const uint32_t loff = ldsBase + (uint32_t)((c * 32 + lane) * 16);
      const uint64_t g =
          (uint64_t)(uintptr_t)(srcRow + (size_t)(c * 32 + lane) * 4);
      asm volatile("global_load_async_to_lds_b128 %0, %1, off"
                   :: "v"(loff), "v"(g)
                   : "memory");
    }
  };

  if (w < B_ROWS) issue_row(w, ldsBase0);            // prologue: stage 0

  int iter = 0;
  for (int r = w; r < B_ROWS; r += totalWaves, ++iter) {
    const int cur = iter & 1;
    const int rn  = r + totalWaves;
    if (rn < B_ROWS) {
      issue_row(rn, cur ? ldsBase0 : ldsBase1);      // prefetch next stage
      // 8 just-issued may stay outstanding; current row's 8 are complete
      asm volatile("s_wait_asynccnt 0x8" ::: "memory");
    } else {
      asm volatile("s_wait_asynccnt 0x0" ::: "memory");
    }

    // ---- post-scatter per-row scalars (wave-uniform) ----
    const bool  isPos    = (r == pos);
    const float age      = isPos ? 0.0f    : (ages_in[r] + dt);
    const float tag      = isPos ? valence : tags_in[r];
    const float strength = isPos ? posStr  : strengths_in[r];

    float af = 1.0f - age * 1.0e-4f;                 // 1 - age/10000
    af = fminf(fmaxf(af, 0.1f), 1.0f);
    const float decay = 0.95f * af * (1.0f + fabsf(tag) * 0.3f);

    // ---- consume staged row from LDS (ds_load_b128), decay, store buf ----
    const float* srcLds = &lds_rows[waveInBlock][cur][0];
    v4f* __restrict__ dstBuf = (v4f*)(bufOut + (size_t)r * COLS);

    v4f v[8];
    float partial = 0.0f;
    #pragma unroll
    for (int c = 0; c < 8; ++c) {
      v4f x = *(const v4f*)(srcLds + (c * 32 + lane) * 4);
      x *= decay;
      v[c] = x;
      partial += fabsf(x[0]) + fabsf(x[1]) + fabsf(x[2]) + fabsf(x[3]);
      __builtin_nontemporal_store(x, &dstBuf[c * 32 + lane]);
    }

    // ---- wave32 abs-sum reduction (no barriers) ----
    float rowsum = partial;
    #pragma unroll
    for (int o = 16; o; o >>= 1) rowsum += __shfl_xor(rowsum, o, 32);

    const bool cand = (strength > 0.5f) && (age > 100.0f) && (rowsum > 0.1f);
    if (lane == 0) candOut[r] = cand ? 1.0f : 0.0f;

    // ---- ltm_ready row: candidate ? decayed row : zeros ----
    // ltm base flat offset 67174403 is only 4B-aligned -> scalar b32 NT stores
    float* __restrict__ lrow = ltmOut + (size_t)r * COLS;
    if (cand) {                                      // wave-uniform branch
      #pragma unroll
      for (int c = 0; c < 8; ++c) {
        const int col = (c * 32 + lane) * 4;
        __builtin_nontemporal_store(v[c][0], lrow + col + 0);
        __builtin_nontemporal_store(v[c][1], lrow + col + 1);
        __builtin_nontemporal_store(v[c][2], lrow + col + 2);
        __builtin_nontemporal_store(v[c][3], lrow + col + 3);
      }
    } else {
      #pragma unroll
      for (int c = 0; c < 8; ++c) {
        const int col = (c * 32 + lane) * 4;
        __builtin_nontemporal_store(0.0f, lrow + col + 0);
        __builtin_nontemporal_store(0.0f, lrow + col + 1);
        __builtin_nontemporal_store(0.0f, lrow + col + 2);
        __builtin_nontemporal_store(0.0f, lrow + col + 3);
      }
    }

    // ---- per-wave scalar accumulation (uniform across lanes) ----
    ageSum  += age;
    occCnt  += (rowsum > 0.01f) ? 1.0f : 0.0f;
    candCnt += cand ? 1.0f : 0.0f;
  }

  if (lane == 0) {
    atomicAdd(out + OUT_OCC,  occCnt * (1.0f / (float)B_ROWS));
    atomicAdd(out + OUT_AGE,  ageSum * (1.0f / (float)B_ROWS));
    atomicAdd(out + OUT_PRES, candCnt);
  }
}

extern "C" void kernel_launch(void* const* d_in, const int* in_sizes, int n_in,
                              void* d_out, int out_size, void* d_ws, size_t ws_size,
                              hipStream_t stream) {
  const float* new_input = (const float*)d_in[0];   // 1024
  const float* membuf    = (const float*)d_in[1];   // 65536*1024
  const float* ages      = (const float*)d_in[2];   // 65536
  const float* tags      = (const float*)d_in[3];   // 65536
  const float* strengths = (const float*)d_in[4];   // 65536
  const int*   wpos      = (const int*)  d_in[5];   // 1
  const float* valence   = (const float*)d_in[6];   // 1
  const float* dt        = (const float*)d_in[7];   // 1
  float* out = (float*)d_out;
  float* ws  = (float*)d_ws;

  stm_init_kernel<<<1, 32, 0, stream>>>(new_input, valence, ws, out);

  // 2048 blocks x 4 waves = 8192 waves; each wave owns 8 rows (grid-stride).
  stm_main_kernel<<<2048, 32 * WAVES_PER_BLOCK, 0, stream>>>(
      new_input, membuf, ages, tags, strengths, wpos, valence, dt, ws, out);
}